// GATNet_26379689132135
// MI455X (gfx1250) — compile-verified
//
#include <hip/hip_runtime.h>
#include <math.h>

#define IN_CH 128
#define HIDC  16
#define HEADS 8
#define OUT_CH 16
#define NEG_SLOPE 0.2f

typedef __attribute__((ext_vector_type(2))) float v2f;
typedef __attribute__((ext_vector_type(8))) float v8f;

// ---- monotonic float<->uint mapping for atomic max on signed floats ----
__device__ __forceinline__ unsigned f2ord(float f) {
    unsigned u = __float_as_uint(f);
    return (u & 0x80000000u) ? ~u : (u | 0x80000000u);
}
__device__ __forceinline__ float ord2f(unsigned u) {
    return __uint_as_float((u & 0x80000000u) ? (u & 0x7FFFFFFFu) : ~u);
}

// ================= WMMA f32 GEMM: D[M,Ncols] = A[M,K] @ B[K,Ncols] =========
// One wave per 16x16 output tile, K stepped by 4 via V_WMMA_F32_16X16X4_F32.
__global__ void gemm_wmma_f32(const float* __restrict__ A,
                              const float* __restrict__ B,
                              float* __restrict__ D,
                              int M, int K, int Ncols) {
    const int lane  = threadIdx.x & 31;
    const int wave  = threadIdx.x >> 5;
    const int tilesN = Ncols >> 4;
    const int tilesM = (M + 15) >> 4;
    const int tile  = blockIdx.x * (blockDim.x >> 5) + wave;
    if (tile >= tilesM * tilesN) return;           // wave-uniform: EXEC stays full
    const int tm = tile / tilesN;
    const int tn = tile - tm * tilesN;

    const int half = lane >> 4;       // 0: lanes 0-15, 1: lanes 16-31
    const int l15  = lane & 15;
    int arow = tm * 16 + l15;
    if (arow >= M) arow = M - 1;                   // clamp A loads (stores guarded)
    const float* ap = A + (size_t)arow * K;
    const int col = tn * 16 + l15;

    v8f acc = {};
    for (int k = 0; k < K; k += 4) {
        const int ka = k + half * 2;
        v2f av, bv;
        av.x = ap[ka];
        av.y = ap[ka + 1];
        bv.x = B[(size_t)ka * Ncols + col];
        bv.y = B[(size_t)(ka + 1) * Ncols + col];
        acc = __builtin_amdgcn_wmma_f32_16x16x4_f32(
            /*neg_a=*/false, av, /*neg_b=*/false, bv,
            /*c_mod=*/(short)0, acc, /*reuse_a=*/false, /*reuse_b=*/false);
    }
    const int mbase = tm * 16 + half * 8;
#pragma unroll
    for (int g = 0; g < 8; ++g) {
        int r = mbase + g;
        if (r < M) D[(size_t)r * Ncols + col] = acc[g];
    }
}

// ====== per-node attention halves: s[n,h]=<h[n,h,:],a_src[h]>, d likewise ===
__global__ void node_att(const float* __restrict__ H,
                         const float* __restrict__ a_src,
                         const float* __restrict__ a_dst,
                         float* __restrict__ s_out, float* __restrict__ d_out_,
                         int Nn, int heads) {
    int i = blockIdx.x * blockDim.x + threadIdx.x;   // i = n*heads + h
    if (i >= Nn * heads) return;
    int h = i % heads;
    const float* hp = H + (size_t)i * 16;            // contiguous [n][h][16]
    const float* as = a_src + h * 16;
    const float* ad = a_dst + h * 16;
    float s = 0.f, d = 0.f;
#pragma unroll
    for (int c = 0; c < 16; ++c) { float v = hp[c]; s = fmaf(v, as[c], s); d = fmaf(v, ad[c], d); }
    s_out[i] = s;
    d_out_[i] = d;
}

__device__ __forceinline__ void edge_sd(const int* ei, int e, int E, int& s, int& d) {
    if (e < E) { s = ei[e]; d = ei[E + e]; }
    else       { s = e - E; d = e - E; }             // self loops appended
}

// =================== segment max of leaky_relu(e) over dst ==================
__global__ void edge_max(const int* __restrict__ ei, int E, int Nn, int heads,
                         const float* __restrict__ as, const float* __restrict__ ad,
                         unsigned* __restrict__ emax) {
    int idx = blockIdx.x * blockDim.x + threadIdx.x;
    if (idx >= (E + Nn) * heads) return;
    int e = idx / heads, h = idx - e * heads;
    int s, d; edge_sd(ei, e, E, s, d);
    float v = as[(size_t)s * heads + h] + ad[(size_t)d * heads + h];
    v = (v > 0.f) ? v : v * NEG_SLOPE;
    atomicMax(&emax[(size_t)d * heads + h], f2ord(v));
}

// =================== segment sum of exp(e - max) over dst ===================
__global__ void edge_denom(const int* __restrict__ ei, int E, int Nn, int heads,
                           const float* __restrict__ as, const float* __restrict__ ad,
                           const unsigned* __restrict__ emax, float* __restrict__ denom) {
    int idx = blockIdx.x * blockDim.x + threadIdx.x;
    if (idx >= (E + Nn) * heads) return;
    int e = idx / heads, h = idx - e * heads;
    int s, d; edge_sd(ei, e, E, s, d);
    float v = as[(size_t)s * heads + h] + ad[(size_t)d * heads + h];
    v = (v > 0.f) ? v : v * NEG_SLOPE;
    float m = ord2f(emax[(size_t)d * heads + h]);
    atomicAdd(&denom[(size_t)d * heads + h], expf(v - m));
}

// ============ alpha-weighted scatter: agg[dst,h,:] += h[src,h,:]*alpha ======
__global__ void edge_scatter(const int* __restrict__ ei, int E, int Nn, int heads,
                             const float* __restrict__ as, const float* __restrict__ ad,
                             const unsigned* __restrict__ emax,
                             const float* __restrict__ denom,
                             const float* __restrict__ Hfeat,
                             float* __restrict__ agg) {
    int idx = blockIdx.x * blockDim.x + threadIdx.x;
    if (idx >= (E + Nn) * heads) return;
    int e = idx / heads, h = idx - e * heads;
    int s, d; edge_sd(ei, e, E, s, d);
    const float* hs = Hfeat + ((size_t)s * heads + h) * 16;
    __builtin_prefetch(hs, 0, 0);                    // global_prefetch_b8
    float v = as[(size_t)s * heads + h] + ad[(size_t)d * heads + h];
    v = (v > 0.f) ? v : v * NEG_SLOPE;
    float m  = ord2f(emax[(size_t)d * heads + h]);
    float al = expf(v - m) / (denom[(size_t)d * heads + h] + 1e-16f);
    float* ag = agg + ((size_t)d * heads + h) * 16;
#pragma unroll
    for (int c = 0; c < 16; ++c) atomicAdd(&ag[c], hs[c] * al);
}

// ===================== out = elu(agg + bias), in place ======================
__global__ void bias_elu(float* __restrict__ agg, const float* __restrict__ b,
                         int Nn, int F) {
    int i = blockIdx.x * blockDim.x + threadIdx.x;
    if (i >= Nn * F) return;
    float v = agg[i] + b[i % F];
    agg[i] = (v > 0.f) ? v : (expf(v) - 1.f);
}

// ============= logits = log_softmax(agg + bias) over 16, in place ===========
__global__ void bias_logsoftmax16(float* __restrict__ logits,
                                  const float* __restrict__ b, int Nn) {
    int n = blockIdx.x * blockDim.x + threadIdx.x;
    if (n >= Nn) return;
    float* p = logits + (size_t)n * 16;
    float v[16];
    float m = -__builtin_inff();
#pragma unroll
    for (int c = 0; c < 16; ++c) { v[c] = p[c] + b[c]; m = fmaxf(m, v[c]); }
    float s = 0.f;
#pragma unroll
    for (int c = 0; c < 16; ++c) s += expf(v[c] - m);
    float lse = m + logf(s);
#pragma unroll
    for (int c = 0; c < 16; ++c) p[c] = v[c] - lse;
}

extern "C" void kernel_launch(void* const* d_in, const int* in_sizes, int n_in,
                              void* d_out, int out_size, void* d_ws, size_t ws_size,
                              hipStream_t stream) {
    const float* x    = (const float*)d_in[0];
    const int*   ei   = (const int*)  d_in[1];
    const float* W1   = (const float*)d_in[2];
    const float* aS1  = (const float*)d_in[3];
    const float* aD1  = (const float*)d_in[4];
    const float* b1   = (const float*)d_in[5];
    const float* W2   = (const float*)d_in[6];
    const float* aS2  = (const float*)d_in[7];
    const float* aD2  = (const float*)d_in[8];
    const float* b2   = (const float*)d_in[9];

    const int Nn = in_sizes[0] / IN_CH;
    const int E  = in_sizes[1] / 2;
    const int ET = E + Nn;                    // with self loops

    float* out    = (float*)d_out;
    float* logits = out;                      // [N,16]  doubles as layer-2 accumulator
    float* emb    = out + (size_t)Nn * OUT_CH;// [N,128] doubles as layer-1 accumulator

    float* ws = (float*)d_ws;
    size_t o = 0;
    float*    h1     = ws + o; o += (size_t)Nn * 128;
    float*    as1    = ws + o; o += (size_t)Nn * HEADS;
    float*    ad1    = ws + o; o += (size_t)Nn * HEADS;
    unsigned* emax1  = (unsigned*)(ws + o); o += (size_t)Nn * HEADS;
    float*    denom1 = ws + o; o += (size_t)Nn * HEADS;
    float*    h2     = ws + o; o += (size_t)Nn * OUT_CH;
    float*    as2    = ws + o; o += (size_t)Nn;
    float*    ad2    = ws + o; o += (size_t)Nn;
    unsigned* emax2  = (unsigned*)(ws + o); o += (size_t)Nn;
    float*    denom2 = ws + o; o += (size_t)Nn;

    // zero all accumulators every call (graph-capture safe, deterministic)
    hipMemsetAsync(emb,    0, (size_t)Nn * 128 * sizeof(float),   stream);
    hipMemsetAsync(logits, 0, (size_t)Nn * OUT_CH * sizeof(float),stream);
    hipMemsetAsync(emax1,  0, (size_t)Nn * HEADS * sizeof(unsigned), stream);
    hipMemsetAsync(denom1, 0, (size_t)Nn * HEADS * sizeof(float), stream);
    hipMemsetAsync(emax2,  0, (size_t)Nn * sizeof(unsigned), stream);
    hipMemsetAsync(denom2, 0, (size_t)Nn * sizeof(float),    stream);

    const int TPB = 256;                      // 8 waves / block (wave32)
    const int WPB = TPB / 32;

    // ---------------- layer 1 ----------------
    {   // h1 = x @ W1   (100000x128x128, fp32 WMMA)
        int tiles = ((Nn + 15) / 16) * (HEADS * HIDC / 16);
        gemm_wmma_f32<<<(tiles + WPB - 1) / WPB, TPB, 0, stream>>>(
            x, W1, h1, Nn, IN_CH, HEADS * HIDC);
    }
    node_att<<<((size_t)Nn * HEADS + TPB - 1) / TPB, TPB, 0, stream>>>(
        h1, aS1, aD1, as1, ad1, Nn, HEADS);

    int eg1 = ((size_t)ET * HEADS + TPB - 1) / TPB;
    edge_max   <<<eg1, TPB, 0, stream>>>(ei, E, Nn, HEADS, as1, ad1, emax1);
    edge_denom <<<eg1, TPB, 0, stream>>>(ei, E, Nn, HEADS, as1, ad1, emax1, denom1);
    edge_scatter<<<eg1, TPB, 0, stream>>>(ei, E, Nn, HEADS, as1, ad1, emax1, denom1,
                                          h1, emb);
    bias_elu<<<((size_t)Nn * 128 + TPB - 1) / TPB, TPB, 0, stream>>>(emb, b1, Nn, 128);

    // ---------------- layer 2 ----------------
    {   // h2 = emb @ W2   (100000x128x16, fp32 WMMA)
        int tiles = ((Nn + 15) / 16) * (OUT_CH / 16);
        gemm_wmma_f32<<<(tiles + WPB - 1) / WPB, TPB, 0, stream>>>(
            emb, W2, h2, Nn, HEADS * HIDC, OUT_CH);
    }
    node_att<<<((size_t)Nn + TPB - 1) / TPB, TPB, 0, stream>>>(
        h2, aS2, aD2, as2, ad2, Nn, 1);

    int eg2 = ((size_t)ET + TPB - 1) / TPB;
    edge_max   <<<eg2, TPB, 0, stream>>>(ei, E, Nn, 1, as2, ad2, emax2);
    edge_denom <<<eg2, TPB, 0, stream>>>(ei, E, Nn, 1, as2, ad2, emax2, denom2);
    edge_scatter<<<eg2, TPB, 0, stream>>>(ei, E, Nn, 1, as2, ad2, emax2, denom2,
                                          h2, logits);
    bias_logsoftmax16<<<(Nn + TPB - 1) / TPB, TPB, 0, stream>>>(logits, b2, Nn);
}